// RotaryPositionalEmbedding_7670811591012
// MI455X (gfx1250) — compile-verified
//
#include <hip/hip_runtime.h>
#include <stdint.h>

// RoPE via block-diagonal structure of R:
//   out[2k]   = c*xe + (-s)*xo        (c, -s) = R[p, 2k, 2k], R[p, 2k, 2k+1]
//   out[2k+1] = c*xo - (-s)*xe
// Bandwidth-bound (~20 MB total, ~1.6 MFLOP) -> no WMMA; use CDNA5 async
// global->LDS DMA to stage the x tile and keep R-gathers as the only
// VGPR-landing loads.

#define DK            128
#define PAIRS         (DK / 2)          // 64 rotation pairs per token
#define TOK_PER_BLOCK 4
#define BLOCK         256               // 8 wave32s; 1 wave = half a token

__global__ __launch_bounds__(BLOCK) void rope_rot_kernel(
    const float* __restrict__ x,        // [ntok, 128]
    const int*   __restrict__ pos,      // [ntok]
    const float* __restrict__ R,        // [2048, 128, 128]
    float*       __restrict__ out,      // [ntok, 128]
    int ntok)
{
    __shared__ float xs[TOK_PER_BLOCK * DK];   // 2 KB x staging tile

    const int t    = threadIdx.x;
    const int tok0 = blockIdx.x * TOK_PER_BLOCK;

    // ---- CDNA5 async DMA: stage this block's x tile (global -> LDS) ----
    // Each of the 256 lanes moves 8 contiguous bytes (fully coalesced).
    {
        const int gElem = tok0 * DK + t * 2;
        if (gElem + 1 < ntok * DK) {
            // Low 32 bits of a generic LDS pointer == wave-relative LDS offset.
            uint32_t lds = (uint32_t)(uintptr_t)(&xs[t * 2]);
            uint32_t off = (uint32_t)(gElem * sizeof(float));
            asm volatile("global_load_async_to_lds_b64 %0, %1, %2"
                         :
                         : "v"(lds), "v"(off), "s"(x)
                         : "memory");
        }
#if __has_builtin(__builtin_amdgcn_s_wait_asynccnt)
        __builtin_amdgcn_s_wait_asynccnt(0);
#else
        asm volatile("s_wait_asynccnt 0x0" ::: "memory");
#endif
    }
    __syncthreads();

    // ---- per-pair rotation ----
    const int localTok = t >> 6;            // 0..3
    const int pair     = t & (PAIRS - 1);   // 0..63
    const int tok      = tok0 + localTok;
    if (tok < ntok) {
        const int p = pos[tok];             // wave-uniform -> scalar load

        // (c, -s) packed adjacently in row 2k of R[p]: stride 2*DK+2 floats.
        const float2 cs = *(const float2*)(R + (size_t)p * (DK * DK)
                                             + (size_t)pair * (2 * DK + 2));
        const float2 xv = *(const float2*)(&xs[localTok * DK + pair * 2]);

        float2 o;
        o.x = fmaf(cs.x, xv.x,  cs.y * xv.y);   // c*xe - s*xo
        o.y = fmaf(cs.x, xv.y, -cs.y * xv.x);   // c*xo + s*xe
        *(float2*)(out + (size_t)tok * DK + pair * 2) = o;
    }
}

extern "C" void kernel_launch(void* const* d_in, const int* in_sizes, int n_in,
                              void* d_out, int out_size, void* d_ws, size_t ws_size,
                              hipStream_t stream)
{
    const float* x   = (const float*)d_in[0];   // [B, S, 128] fp32
    const int*   pos = (const int*)  d_in[1];   // [B, S] integer
    const float* R   = (const float*)d_in[2];   // [2048, 128, 128] fp32
    float*       out = (float*)d_out;

    const int ntok = in_sizes[0] / DK;          // B*S
    const int grid = (ntok + TOK_PER_BLOCK - 1) / TOK_PER_BLOCK;

    rope_rot_kernel<<<grid, BLOCK, 0, stream>>>(x, pos, R, out, ntok);
}